// MultiHeadAttention_41094247088139
// MI455X (gfx1250) — compile-verified
//
#include <hip/hip_runtime.h>
#include <math.h>

// Problem constants (match reference)
#define BQ 4
#define LQN 256
#define LKN 256
#define DD 512
#define HH 8
#define DHD 64
#define AHN 40
#define BHN (BQ*HH)   // 32

typedef __attribute__((ext_vector_type(2))) float v2f;
typedef __attribute__((ext_vector_type(4))) float v4f;
typedef __attribute__((ext_vector_type(8))) float v8f;

#if __has_builtin(__builtin_amdgcn_tanhf)
__device__ __forceinline__ float fast_tanh(float x) { return __builtin_amdgcn_tanhf(x); }
#else
__device__ __forceinline__ float fast_tanh(float x) { return tanhf(x); }
#endif

#define WMMA_F32(acc, a, b) \
  acc = __builtin_amdgcn_wmma_f32_16x16x4_f32(false, a, false, b, (short)0, acc, false, false)

// ---------------------------------------------------------------------------
// Kernel 1: fused Q/K/V projection.  C[m,n] = sum_k X[m,k] * W[n,k]
// 2x2 register-blocked WMMA-f32: one wave owns a 32x32 output block
// (4 accumulators); each K-step loads 2 A + 2 B fragments feeding 4 WMMAs,
// halving L2 traffic vs one-tile-per-wave.
// f32 WMMA fragment layout (ISA 7.12.2):
//   A 16x4 : lane t -> m = t%16, k0 = 2*(t/16);  a = {A[m][k0], A[m][k0+1]}
//   B 4x16 : lane t -> n = t%16, k0 = 2*(t/16);  b = {B[k0][n], B[k0+1][n]} = W row n
//   C/D    : lane t, vgpr i -> D[i + 8*(t/16)][t%16]
// Q/K stored head-split (BH, L, DH); V stored transposed (BH, DH, LK).
// ---------------------------------------------------------------------------
__global__ __launch_bounds__(256) void qkv_proj_kernel(
    const float* __restrict__ q_in, const float* __restrict__ k_in,
    const float* __restrict__ v_in,
    const float* __restrict__ Wq, const float* __restrict__ Wk,
    const float* __restrict__ Wv,
    float* __restrict__ qp, float* __restrict__ kp, float* __restrict__ vpt)
{
  const unsigned tid  = threadIdx.x;
  const unsigned lane = tid & 31u;
  const unsigned wave = tid >> 5;
  unsigned job = blockIdx.x * 8u + wave;           // 3 * 32 * 16 = 1536 jobs
  const unsigned mat = job / (32u * 16u);
  job %= (32u * 16u);
  const unsigned mt = job / 16u;                   // 32-row block
  const unsigned nt = job % 16u;                   // 32-col block

  const float* A = (mat == 0) ? q_in : (mat == 1) ? k_in : v_in;
  const float* W = (mat == 0) ? Wq   : (mat == 1) ? Wk   : Wv;

  const unsigned l15 = lane & 15u;
  const unsigned kh  = (lane >> 4) * 2u;
  const unsigned mr0 = mt * 32u + l15;
  const unsigned nr0 = nt * 32u + l15;

  const float* arow0 = A + (size_t)mr0 * DD + kh;
  const float* arow1 = arow0 + (size_t)16 * DD;
  const float* brow0 = W + (size_t)nr0 * DD + kh;
  const float* brow1 = brow0 + (size_t)16 * DD;

  v8f acc00 = {}, acc01 = {}, acc10 = {}, acc11 = {};
  #pragma unroll 4
  for (int k = 0; k < DD; k += 4) {
    v2f a0 = *(const v2f*)(arow0 + k);
    v2f a1 = *(const v2f*)(arow1 + k);
    v2f b0 = *(const v2f*)(brow0 + k);
    v2f b1 = *(const v2f*)(brow1 + k);
    WMMA_F32(acc00, a0, b0);
    WMMA_F32(acc01, a0, b1);
    WMMA_F32(acc10, a1, b0);
    WMMA_F32(acc11, a1, b1);
  }

  const unsigned moff = 8u * (lane >> 4);

  #pragma unroll
  for (int mi = 0; mi < 2; ++mi) {
    #pragma unroll
    for (int ni = 0; ni < 2; ++ni) {
      const v8f& acc = (mi == 0) ? (ni == 0 ? acc00 : acc01)
                                 : (ni == 0 ? acc10 : acc11);
      const unsigned n  = nt * 32u + ni * 16u + l15;   // output feature
      const unsigned h  = n >> 6;                      // head
      const unsigned dh = n & 63u;
      const unsigned row0 = mt * 32u + mi * 16u + moff; // = b*256 + l (8-aligned)
      const unsigned bb = row0 >> 8;
      const unsigned l0 = row0 & 255u;
      if (mat < 2) {
        float* out = (mat == 0) ? qp : kp;
        float* dst = out + ((size_t)(bb * HH + h) * LQN + l0) * DHD + dh;
        #pragma unroll
        for (int i = 0; i < 8; ++i) dst[(size_t)i * DHD] = acc[i];
      } else {
        // transposed V: vpt[(bh*DH + dh)*LK + l]; 8 consecutive l per lane
        float* dst = vpt + ((size_t)(bb * HH + h) * DHD + dh) * LKN + l0;
        v4f lo = { acc[0], acc[1], acc[2], acc[3] };
        v4f hi = { acc[4], acc[5], acc[6], acc[7] };
        *(v4f*)(dst)     = lo;
        *(v4f*)(dst + 4) = hi;
      }
    }
  }
}

// ---------------------------------------------------------------------------
// Kernel 2: additive-attention features  qf = q @ Aq^T, kf = k @ Ak^T
// M = BH*L = 8192, K = 64, N = 40 padded to 48 (guarded).
// ---------------------------------------------------------------------------
__global__ __launch_bounds__(256) void addfeat_kernel(
    const float* __restrict__ qp, const float* __restrict__ kp,
    const float* __restrict__ Aq, const float* __restrict__ Ak,
    float* __restrict__ qf, float* __restrict__ kf)
{
  const unsigned tid  = threadIdx.x;
  const unsigned lane = tid & 31u;
  const unsigned wave = tid >> 5;
  unsigned tile = blockIdx.x * 8u + wave;          // 2 * 512 * 3 = 3072 tiles
  const unsigned mat = tile / (512u * 3u);
  tile %= (512u * 3u);
  const unsigned mt = tile / 3u;
  const unsigned nt = tile % 3u;

  const float* A  = mat ? kp : qp;
  const float* Wf = mat ? Ak : Aq;
  float* out      = mat ? kf : qf;

  const unsigned mr = mt * 16u + (lane & 15u);
  const unsigned nr = nt * 16u + (lane & 15u);
  const unsigned kh = (lane >> 4) * 2u;
  const float* arow = A  + (size_t)mr * DHD + kh;
  const float* brow = Wf + (size_t)nr * DHD + kh;
  const bool nvalid = (nr < AHN);

  v8f acc = {};
  #pragma unroll
  for (int k = 0; k < DHD; k += 4) {
    v2f a = *(const v2f*)(arow + k);
    v2f b = { 0.f, 0.f };
    if (nvalid) b = *(const v2f*)(brow + k);
    WMMA_F32(acc, a, b);
  }
  if (nvalid) {
    const unsigned moff = 8u * (lane >> 4);
    #pragma unroll
    for (int i = 0; i < 8; ++i) {
      unsigned row = mt * 16u + moff + i;
      out[(size_t)row * AHN + nr] = acc[i];
    }
  }
}

// ---------------------------------------------------------------------------
// Kernel 3: fused additive scores (tanh) + softmax + w@v (WMMA).
// One block per (bh, 16-query tile).  16x256 weight tile lives in LDS.
// ---------------------------------------------------------------------------
__global__ __launch_bounds__(256) void attn_kernel(
    const float* __restrict__ qf, const float* __restrict__ kf,
    const float* __restrict__ vpt, const float* __restrict__ av,
    float* __restrict__ attn_out)
{
  __shared__ float sc[16][LKN];    // 16 KB: scores -> softmax weights
  __shared__ float qfs[16][AHN];
  __shared__ float avs[AHN];
  __shared__ float red[16][16];
  __shared__ float red2[16];

  const unsigned tid = threadIdx.x;
  const unsigned bh  = blockIdx.x >> 4;    // gridDim.x = 32 * 16
  const unsigned qt  = blockIdx.x & 15u;

  for (unsigned idx = tid; idx < 16u * AHN; idx += 256u) {
    unsigned q = idx / AHN, h = idx % AHN;
    qfs[q][h] = qf[((size_t)bh * LQN + qt * 16u + q) * AHN + h];
  }
  if (tid < AHN) avs[tid] = av[tid];
  __syncthreads();

  const unsigned q = tid >> 4;
  const unsigned j = tid & 15u;

  float qr[AHN];
  #pragma unroll
  for (int h = 0; h < AHN; ++h) qr[h] = qfs[q][h];

  // scores[q][k] = sum_h av[h] * tanh(qf[q][h] + kf[k][h])
  for (unsigned t = 0; t < 16u; ++t) {
    unsigned k = j + 16u * t;
    const float* kr = kf + ((size_t)bh * LKN + k) * AHN;
    float s = 0.f;
    #pragma unroll
    for (int h = 0; h < AHN; ++h)
      s += avs[h] * fast_tanh(qr[h] + kr[h]);
    sc[q][k] = s;
  }
  __syncthreads();

  // row softmax over 256 keys (16 threads per row)
  float m = -3.4e38f;
  for (unsigned t = 0; t < 16u; ++t) m = fmaxf(m, sc[q][j + 16u * t]);
  red[q][j] = m;
  __syncthreads();
  if (j == 0) {
    float rm = red[q][0];
    for (int t = 1; t < 16; ++t) rm = fmaxf(rm, red[q][t]);
    red2[q] = rm;
  }
  __syncthreads();
  const float rm = red2[q];
  float s = 0.f;
  for (unsigned t = 0; t < 16u; ++t) {
    float e = __expf(sc[q][j + 16u * t] - rm);
    sc[q][j + 16u * t] = e;
    s += e;
  }
  red[q][j] = s;
  __syncthreads();
  if (j == 0) {
    float rs = 0.f;
    for (int t = 0; t < 16; ++t) rs += red[q][t];
    red2[q] = 1.0f / rs;
  }
  __syncthreads();
  const float inv = red2[q];
  for (unsigned t = 0; t < 16u; ++t) sc[q][j + 16u * t] *= inv;
  __syncthreads();

  // out(16x64) = w(16x256) @ v(256x64); waves 0..3 each own one 16-col tile
  const unsigned wave = tid >> 5;
  const unsigned lane = tid & 31u;
  if (wave < 4) {
    const unsigned nloc = wave * 16u + (lane & 15u);       // dh 0..63
    const unsigned kh   = (lane >> 4) * 2u;
    const float* brow = vpt + ((size_t)bh * DHD + nloc) * LKN + kh;
    const float* awrow = &sc[lane & 15u][kh];
    v8f acc = {};
    #pragma unroll 8
    for (int k = 0; k < LKN; k += 4) {
      v2f a = *(const v2f*)(awrow + k);    // ds_load_b64 from LDS
      v2f b = *(const v2f*)(brow + k);     // contiguous thanks to V transpose
      WMMA_F32(acc, a, b);
    }
    const unsigned bb = bh >> 3, hh = bh & 7u;
    const unsigned moff = 8u * (lane >> 4);
    #pragma unroll
    for (int i = 0; i < 8; ++i) {
      unsigned l = qt * 16u + moff + i;
      attn_out[((size_t)bb * LQN + l) * DD + hh * DHD + nloc] = acc[i];
    }
  }
}

// ---------------------------------------------------------------------------
// Kernel 4: output projection  out = attn @ Wo^T   (1024 x 512 x 512)
// Same 2x2 register-blocked WMMA scheme as kernel 1.
// ---------------------------------------------------------------------------
__global__ __launch_bounds__(256) void out_proj_kernel(
    const float* __restrict__ attn, const float* __restrict__ Wo,
    float* __restrict__ out)
{
  const unsigned tid  = threadIdx.x;
  const unsigned lane = tid & 31u;
  const unsigned wave = tid >> 5;
  unsigned job = blockIdx.x * 8u + wave;     // 32 * 16 = 512 jobs
  const unsigned mt = job / 16u;
  const unsigned nt = job % 16u;

  const unsigned l15 = lane & 15u;
  const unsigned kh  = (lane >> 4) * 2u;
  const unsigned mr0 = mt * 32u + l15;
  const unsigned nr0 = nt * 32u + l15;

  const float* arow0 = attn + (size_t)mr0 * DD + kh;
  const float* arow1 = arow0 + (size_t)16 * DD;
  const float* brow0 = Wo + (size_t)nr0 * DD + kh;
  const float* brow1 = brow0 + (size_t)16 * DD;

  v8f acc00 = {}, acc01 = {}, acc10 = {}, acc11 = {};
  #pragma unroll 4
  for (int k = 0; k < DD; k += 4) {
    v2f a0 = *(const v2f*)(arow0 + k);
    v2f a1 = *(const v2f*)(arow1 + k);
    v2f b0 = *(const v2f*)(brow0 + k);
    v2f b1 = *(const v2f*)(brow1 + k);
    WMMA_F32(acc00, a0, b0);
    WMMA_F32(acc01, a0, b1);
    WMMA_F32(acc10, a1, b0);
    WMMA_F32(acc11, a1, b1);
  }

  const unsigned moff = 8u * (lane >> 4);
  #pragma unroll
  for (int mi = 0; mi < 2; ++mi) {
    #pragma unroll
    for (int ni = 0; ni < 2; ++ni) {
      const v8f& acc = (mi == 0) ? (ni == 0 ? acc00 : acc01)
                                 : (ni == 0 ? acc10 : acc11);
      const unsigned nr = nt * 32u + ni * 16u + l15;
      float* dst = out + (size_t)(mt * 32u + mi * 16u + moff) * DD + nr;
      #pragma unroll
      for (int i = 0; i < 8; ++i) dst[(size_t)i * DD] = acc[i];
    }
  }
}

// ---------------------------------------------------------------------------
extern "C" void kernel_launch(void* const* d_in, const int* in_sizes, int n_in,
                              void* d_out, int out_size, void* d_ws, size_t ws_size,
                              hipStream_t stream) {
  (void)in_sizes; (void)n_in; (void)out_size; (void)ws_size;
  const float* queries = (const float*)d_in[0];
  const float* keys    = (const float*)d_in[1];
  const float* values  = (const float*)d_in[2];
  const float* Wq      = (const float*)d_in[3];
  const float* Wk      = (const float*)d_in[4];
  const float* Wv      = (const float*)d_in[5];
  const float* Wo      = (const float*)d_in[6];
  const float* Aq      = (const float*)d_in[7];
  const float* Ak      = (const float*)d_in[8];
  const float* av      = (const float*)d_in[9];

  float* ws   = (float*)d_ws;
  float* qp   = ws;                                   // (BH, LQ, DH)
  float* kp   = qp  + (size_t)BHN * LQN * DHD;        // (BH, LK, DH)
  float* vpt  = kp  + (size_t)BHN * LKN * DHD;        // (BH, DH, LK)  transposed
  float* qf   = vpt + (size_t)BHN * DHD * LKN;        // (BH, LQ, AH)
  float* kf   = qf  + (size_t)BHN * LQN * AHN;        // (BH, LK, AH)
  float* attn = kf  + (size_t)BHN * LKN * AHN;        // (B, LQ, D)

  qkv_proj_kernel<<<192, 256, 0, stream>>>(queries, keys, values,
                                           Wq, Wk, Wv, qp, kp, vpt);
  addfeat_kernel<<<384, 256, 0, stream>>>(qp, kp, Aq, Ak, qf, kf);
  attn_kernel<<<512, 256, 0, stream>>>(qf, kf, vpt, av, attn);
  out_proj_kernel<<<64, 256, 0, stream>>>(attn, Wo, (float*)d_out);
}